// PureCartesianTransformerLayer_19172734009894
// MI455X (gfx1250) — compile-verified
//
#include <hip/hip_runtime.h>
#include <math.h>

// ---------------------------------------------------------------------------
// CDNA5 (gfx1250) implementation of the equivariant GNN layer.
// All GEMMs use V_WMMA_F32_16X16X4_F32 (full fp32 precision).
// ---------------------------------------------------------------------------

typedef __attribute__((ext_vector_type(2))) float v2f;
typedef __attribute__((ext_vector_type(8))) float v8f;

__device__ __forceinline__ float silu_f(float x) { return x / (1.0f + expf(-x)); }

__device__ __forceinline__ v8f wmma4(v2f a, v2f b, v8f c) {
  // D(16x16,f32) = A(16x4,f32) * B(4x16,f32) + C
  return __builtin_amdgcn_wmma_f32_16x16x4_f32(false, a, false, b, (short)0, c,
                                               false, false);
}

// wave-private LDS staging: in-order DS + explicit dscnt wait gives cross-lane
// visibility without any workgroup barrier.
__device__ __forceinline__ void lds_wait() {
  asm volatile("s_wait_dscnt 0x0" ::: "memory");
}

// ---------------------------------------------------------------------------
// K1: per-edge geometry + Gaussian radial basis + in-degree histogram
// ---------------------------------------------------------------------------
__global__ __launch_bounds__(256) void k1_geom(
    const float* __restrict__ pos, const float* __restrict__ eshift,
    const float* __restrict__ cell, const int* __restrict__ batchArr,
    const int* __restrict__ esrc, const int* __restrict__ edst,
    float* __restrict__ ubuf, float* __restrict__ e2buf,
    float* __restrict__ embbuf, float* __restrict__ deg, int E) {
  int e = blockIdx.x * 256 + threadIdx.x;
  if (e >= E) return;
  int s = esrc[e], d = edst[e];
  int bs = batchArr[s];
  const float* C = cell + bs * 9;
  float s0 = eshift[e * 3 + 0], s1 = eshift[e * 3 + 1], s2 = eshift[e * 3 + 2];
  float v[3];
#pragma unroll
  for (int j = 0; j < 3; ++j)
    v[j] = pos[d * 3 + j] - pos[s * 3 + j] + s0 * C[0 + j] + s1 * C[3 + j] +
           s2 * C[6 + j];
  float len = sqrtf(v[0] * v[0] + v[1] * v[1] + v[2] * v[2]);
  float inv = 1.0f / fmaxf(len, 1e-9f);
  float u[3];
#pragma unroll
  for (int j = 0; j < 3; ++j) {
    u[j] = v[j] * inv;
    ubuf[e * 3 + j] = u[j];
  }
#pragma unroll
  for (int i = 0; i < 3; ++i)
#pragma unroll
    for (int j = 0; j < 3; ++j) e2buf[e * 9 + i * 3 + j] = u[i] * u[j];
  const float step = 6.0f / 17.0f;
  const float scale = 4.0f / 1.12f;  // sqrt(NBASIS)/1.12
#pragma unroll
  for (int b = 0; b < 16; ++b) {
    float dd = (len - (float)(b + 1) * step) / step;
    embbuf[e * 16 + b] = expf(-dd * dd) * scale;
  }
  atomicAdd(&deg[d], 1.0f);
}

// ---------------------------------------------------------------------------
// K2: atom MLP  Ai = (emb_table[A] @ W0 + b0).silu @ W1 + b1   (N x 8)
// ---------------------------------------------------------------------------
__global__ __launch_bounds__(256) void k2_atom(
    const float* __restrict__ emb_table, const int* __restrict__ Aarr,
    const float* __restrict__ w0, const float* __restrict__ b0,
    const float* __restrict__ w1, const float* __restrict__ b1,
    float* __restrict__ Ai, int N) {
  int n = blockIdx.x * 256 + threadIdx.x;
  if (n >= N) return;
  const float* x = emb_table + Aarr[n] * 16;
  float h[64];
#pragma unroll 4
  for (int j = 0; j < 64; ++j) {
    float t = b0[j];
    for (int i = 0; i < 16; ++i) t += x[i] * w0[i * 64 + j];
    h[j] = silu_f(t);
  }
#pragma unroll
  for (int o = 0; o < 8; ++o) {
    float t = b1[o];
    for (int j = 0; j < 64; ++j) t += h[j] * w1[j * 8 + o];
    Ai[n * 8 + o] = t;
  }
}

// ---------------------------------------------------------------------------
// shared trunk: X(16x16 LDS) -> H(16x64 LDS), two silu layers, WMMA f32
// ---------------------------------------------------------------------------
__device__ __forceinline__ void edge_trunk(const float* X, float* H,
                                           const float* w0, const float* b0,
                                           const float* w1, const float* b1,
                                           int lane) {
  const int m = lane & 15;
  const int khi = (lane >= 16) ? 2 : 0;
  const int mhi = (lane >= 16) ? 8 : 0;
  const int ncol = lane & 15;
  // layer 1: 16 -> 64
  for (int nt = 0; nt < 4; ++nt) {
    v8f acc = {};
    const int col = nt * 16 + ncol;
#pragma unroll
    for (int kk = 0; kk < 4; ++kk) {
      const int k0 = 4 * kk + khi;
      v2f a = *(const v2f*)(X + m * 16 + k0);
      v2f b;
      b.x = w0[k0 * 64 + col];
      b.y = w0[(k0 + 1) * 64 + col];
      acc = wmma4(a, b, acc);
    }
    float bb = b0[col];
#pragma unroll
    for (int j = 0; j < 8; ++j) H[(j + mhi) * 64 + col] = silu_f(acc[j] + bb);
  }
  lds_wait();
  // layer 2: 64 -> 64 (register-staged, then overwrite H)
  v8f acc2[4];
  for (int nt = 0; nt < 4; ++nt) {
    v8f acc = {};
    const int col = nt * 16 + ncol;
#pragma unroll
    for (int kk = 0; kk < 16; ++kk) {
      const int k0 = 4 * kk + khi;
      v2f a = *(const v2f*)(H + m * 64 + k0);
      v2f b;
      b.x = w1[k0 * 64 + col];
      b.y = w1[(k0 + 1) * 64 + col];
      acc = wmma4(a, b, acc);
    }
    acc2[nt] = acc;
  }
  lds_wait();
  for (int nt = 0; nt < 4; ++nt) {
    const int col = nt * 16 + ncol;
    float bb = b1[col];
#pragma unroll
    for (int j = 0; j < 8; ++j)
      H[(j + mhi) * 64 + col] = silu_f(acc2[nt][j] + bb);
  }
  lds_wait();
}

// ---------------------------------------------------------------------------
// K3: fc1 trunk + final layer (first 1536 cols) fused with first tensor
// product contraction s[p,o] = sum_{c,d} w1[p,o,c,d]*Asrc[c]*Adst[d] and
// scatter into f1raw.  One wave per 16-edge tile.
// per-wave LDS: X 256 | H 1024 | OP 1024 | S 384  = 2688 floats
// ---------------------------------------------------------------------------
__global__ __launch_bounds__(128) void k3_fc1(
    const float* __restrict__ embbuf, const float* __restrict__ Ai,
    const int* __restrict__ esrc, const int* __restrict__ edst,
    const float* __restrict__ ubuf, const float* __restrict__ e2buf,
    const float* __restrict__ w0, const float* __restrict__ b0,
    const float* __restrict__ w1, const float* __restrict__ b1,
    const float* __restrict__ w2, const float* __restrict__ b2,
    float* __restrict__ f1raw, int E) {
  __shared__ __align__(16) float sm[4 * 2688];
  const int wid = threadIdx.x >> 5;
  const int lane = threadIdx.x & 31;
  const int tile = blockIdx.x * 4 + wid;
  const int e0 = tile * 16;
  if (e0 >= E) return;
  float* X = sm + wid * 2688;
  float* H = X + 256;
  float* OP = H + 1024;
  float* S = OP + 1024;

  for (int idx = lane; idx < 256; idx += 32) {
    int r = idx >> 4, c = idx & 15;
    X[idx] = (e0 + r < E) ? embbuf[(e0 + r) * 16 + c] : 0.f;
  }
  for (int idx = lane; idx < 1024; idx += 32) {
    int r = idx >> 6, q = idx & 63;
    float v = 0.f;
    if (e0 + r < E) {
      int c = q >> 3, d = q & 7;
      v = Ai[esrc[e0 + r] * 8 + c] * Ai[edst[e0 + r] * 8 + d];
    }
    OP[idx] = v;
  }
  lds_wait();
  edge_trunk(X, H, w0, b0, w1, b1, lane);

  const int m = lane & 15;
  const int khi = (lane >= 16) ? 2 : 0;
  const int mhi = (lane >= 16) ? 8 : 0;
  const int ncol = lane & 15;
  // 24 (p,o) blocks of 64 columns each; only p<3 columns of fc1's final layer
  for (int bpo = 0; bpo < 24; ++bpo) {
    float sacc[8] = {0.f, 0.f, 0.f, 0.f, 0.f, 0.f, 0.f, 0.f};
    for (int t = 0; t < 4; ++t) {
      v8f acc = {};
      const int col = bpo * 64 + t * 16 + ncol;
#pragma unroll
      for (int kk = 0; kk < 16; ++kk) {
        const int k0 = 4 * kk + khi;
        v2f a = *(const v2f*)(H + m * 64 + k0);
        v2f b;
        b.x = w2[k0 * 3584 + col];
        b.y = w2[(k0 + 1) * 3584 + col];
        acc = wmma4(a, b, acc);
      }
      const float bb = b2[col];
      const int q = t * 16 + ncol;  // local (c,d) column within block
#pragma unroll
      for (int j = 0; j < 8; ++j) {
        float v = (acc[j] + bb) * OP[(j + mhi) * 64 + q];
        v += __shfl_xor(v, 1, 32);
        v += __shfl_xor(v, 2, 32);
        v += __shfl_xor(v, 4, 32);
        v += __shfl_xor(v, 8, 32);
        sacc[j] += v;
      }
    }
    if (ncol == 0) {
#pragma unroll
      for (int j = 0; j < 8; ++j) S[(j + mhi) * 24 + bpo] = sacc[j];
    }
  }
  lds_wait();
  // scatter: f1 contributions = s[p,o] * e_lo[i]
  for (int idx = lane; idx < 16 * 104; idx += 32) {
    int r = idx / 104, slot = idx - r * 104;
    if (e0 + r >= E) continue;
    int dst = edst[e0 + r];
    float val;
    int o, off;
    if (slot < 8) {
      o = slot;
      off = 0;
      val = S[r * 24 + o];
    } else if (slot < 32) {
      int s2 = slot - 8;
      o = s2 / 3;
      int i = s2 - o * 3;
      off = 1 + i;
      val = S[r * 24 + 8 + o] * ubuf[(e0 + r) * 3 + i];
    } else {
      int s2 = slot - 32;
      o = s2 / 9;
      int i = s2 - o * 9;
      off = 4 + i;
      val = S[r * 24 + 16 + o] * e2buf[(e0 + r) * 9 + i];
    }
    atomicAdd(&f1raw[dst * 104 + o * 13 + off], val);
  }
}

// ---------------------------------------------------------------------------
// K4: f1 = f1raw / max(deg,1)   (element-wise over N*104)
// ---------------------------------------------------------------------------
__global__ __launch_bounds__(256) void k4_mean(float* __restrict__ f1,
                                               const float* __restrict__ deg,
                                               int total) {
  int idx = blockIdx.x * 256 + threadIdx.x;
  if (idx >= total) return;
  int n = idx / 104;
  f1[idx] = f1[idx] / fmaxf(deg[n], 1.0f);
}

// ---------------------------------------------------------------------------
// K5: fc2 trunk + final layer (448 cols) fused with all 7 paths of the 2nd
// tensor product using g1 = f1[src].  One wave per 16-edge tile.
// per-wave LDS: X 256 | H 1024 | G 1664 | Q 384 | S2 1664 | U 48 | E2 144
//             = 5184 floats
// ---------------------------------------------------------------------------
__global__ __launch_bounds__(64) void k5_fc2(
    const float* __restrict__ embbuf, const float* __restrict__ f1,
    const int* __restrict__ esrc, const int* __restrict__ edst,
    const float* __restrict__ ubuf, const float* __restrict__ e2buf,
    const float* __restrict__ w0, const float* __restrict__ b0,
    const float* __restrict__ w1, const float* __restrict__ b1,
    const float* __restrict__ w2, const float* __restrict__ b2,
    float* __restrict__ f2raw, int E) {
  __shared__ __align__(16) float sm[2 * 5184];
  const int wid = threadIdx.x >> 5;
  const int lane = threadIdx.x & 31;
  const int tile = blockIdx.x * 2 + wid;
  const int e0 = tile * 16;
  if (e0 >= E) return;
  float* X = sm + wid * 5184;
  float* H = X + 256;
  float* G = H + 1024;   // g1 = f1[src], [16][8][13]
  float* Q = G + 1664;   // [16][24]: q0 | qd1 | qd2
  float* S2 = Q + 384;   // per-edge f2 accumulators [16][104]
  float* U = S2 + 1664;  // [16][3]
  float* E2v = U + 48;   // [16][9]

  for (int idx = lane; idx < 256; idx += 32) {
    int r = idx >> 4, c = idx & 15;
    X[idx] = (e0 + r < E) ? embbuf[(e0 + r) * 16 + c] : 0.f;
  }
  for (int idx = lane; idx < 48; idx += 32) {
    int r = idx / 3, i = idx - r * 3;
    U[idx] = (e0 + r < E) ? ubuf[(e0 + r) * 3 + i] : 0.f;
  }
  for (int idx = lane; idx < 144; idx += 32) {
    int r = idx / 9, i = idx - r * 9;
    E2v[idx] = (e0 + r < E) ? e2buf[(e0 + r) * 9 + i] : 0.f;
  }
  for (int idx = lane; idx < 1664; idx += 32) {
    int r = idx / 104, k = idx - r * 104;
    G[idx] = (e0 + r < E) ? f1[esrc[e0 + r] * 104 + k] : 0.f;
    S2[idx] = 0.f;
  }
  lds_wait();
  // reduced right-hand vectors
  for (int idx = lane; idx < 128; idx += 32) {
    int r = idx >> 3, c = idx & 7;
    Q[r * 24 + c] = G[r * 104 + c * 13];
    float q1 = 0.f, q2 = 0.f;
#pragma unroll
    for (int i = 0; i < 3; ++i) q1 += G[r * 104 + c * 13 + 1 + i] * U[r * 3 + i];
#pragma unroll
    for (int i = 0; i < 9; ++i) q2 += G[r * 104 + c * 13 + 4 + i] * E2v[r * 9 + i];
    Q[r * 24 + 8 + c] = q1;
    Q[r * 24 + 16 + c] = q2;
  }
  lds_wait();
  edge_trunk(X, H, w0, b0, w1, b1, lane);

  const int m = lane & 15;
  const int khi = (lane >= 16) ? 2 : 0;
  const int mhi = (lane >= 16) ? 8 : 0;
  const int ncol = lane & 15;
  const int c = ncol & 7;
  const int osel = ncol >> 3;
  const bool isred = (c == 0);
  const float INV_SQRT3 = 0.57735026918962576f;

  for (int t = 0; t < 28; ++t) {
    const int p = t >> 2;               // path index, uniform across wave
    const int o = (t & 3) * 2 + osel;   // output channel
    v8f acc = {};
    const int col = t * 16 + ncol;
#pragma unroll
    for (int kk = 0; kk < 16; ++kk) {
      const int k0 = 4 * kk + khi;
      v2f a = *(const v2f*)(H + m * 64 + k0);
      v2f b;
      b.x = w2[k0 * 448 + col];
      b.y = w2[(k0 + 1) * 448 + col];
      acc = wmma4(a, b, acc);
    }
    const float bb = b2[col];
#pragma unroll
    for (int j = 0; j < 8; ++j) {
      const int mm = j + mhi;
      const float w = acc[j] + bb;
      if (p == 1) {  // (1,0,1): sum_c w2*g1_1[c,i] -> L1
        for (int i = 0; i < 3; ++i) {
          float v = w * G[mm * 104 + c * 13 + 1 + i];
          v += __shfl_xor(v, 1, 32);
          v += __shfl_xor(v, 2, 32);
          v += __shfl_xor(v, 4, 32);
          if (isred) S2[mm * 104 + o * 13 + 1 + i] += v;
        }
      } else if (p == 2) {  // (2,0,2): sum_c w2*g1_2[c,i] -> L2
        for (int i = 0; i < 9; ++i) {
          float v = w * G[mm * 104 + c * 13 + 4 + i];
          v += __shfl_xor(v, 1, 32);
          v += __shfl_xor(v, 2, 32);
          v += __shfl_xor(v, 4, 32);
          if (isred) S2[mm * 104 + o * 13 + 4 + i] += v;
        }
      } else {
        float V;
        if (p == 5)
          V = Q[mm * 24 + 8 + c];       // qd1
        else if (p == 6)
          V = Q[mm * 24 + 16 + c];      // qd2
        else
          V = Q[mm * 24 + c];           // q0 (p=0,3,4)
        float v = w * V;
        v += __shfl_xor(v, 1, 32);
        v += __shfl_xor(v, 2, 32);
        v += __shfl_xor(v, 4, 32);
        if (isred) {
          if (p == 0)
            S2[mm * 104 + o * 13] += v;
          else if (p == 3) {
            for (int i = 0; i < 3; ++i)
              S2[mm * 104 + o * 13 + 1 + i] += v * U[mm * 3 + i];
          } else if (p == 4) {
            for (int i = 0; i < 9; ++i)
              S2[mm * 104 + o * 13 + 4 + i] += v * E2v[mm * 9 + i];
          } else if (p == 5)
            S2[mm * 104 + o * 13] += v * INV_SQRT3;
          else
            S2[mm * 104 + o * 13] += v * (1.0f / 3.0f);
        }
      }
    }
  }
  lds_wait();
  for (int idx = lane; idx < 1664; idx += 32) {
    int r = idx / 104, k = idx - r * 104;
    if (e0 + r >= E) continue;
    atomicAdd(&f2raw[edst[e0 + r] * 104 + k], S2[idx]);
  }
}

// ---------------------------------------------------------------------------
// K6a: per-node invariants + gram/W_inv -> feats (N x 128)
// ---------------------------------------------------------------------------
__global__ __launch_bounds__(256) void k6a_feats(
    const float* __restrict__ f1, const float* __restrict__ f2raw,
    const float* __restrict__ deg, const float* __restrict__ Winv0,
    const float* __restrict__ Winv1, const float* __restrict__ Winv2,
    float* __restrict__ feats, int N) {
  int n = blockIdx.x * 256 + threadIdx.x;
  if (n >= N) return;
  float inv_dg = 1.0f / fmaxf(deg[n], 1.0f);
  float f1v[104], f2v[104];
  for (int k = 0; k < 104; ++k) {
    f1v[k] = f1[n * 104 + k];
    f2v[k] = f2raw[n * 104 + k] * inv_dg;
  }
  float* F = feats + n * 128;
  const float s1 = 0.57735026918962576f, s2 = 1.0f / 3.0f;
  for (int o = 0; o < 8; ++o) {
    float a0 = f1v[o * 13];
    F[o] = a0 * a0;
    F[8 + o] = 0.f;
    float t = 0.f;
    for (int i = 1; i < 4; ++i) { float x = f1v[o * 13 + i]; t += x * x; }
    F[16 + o] = t * s1;
    F[24 + o] = 0.f;
    t = 0.f;
    for (int i = 4; i < 13; ++i) { float x = f1v[o * 13 + i]; t += x * x; }
    F[32 + o] = t * s2;
    F[40 + o] = 0.f;
    a0 = f2v[o * 13];
    F[48 + o] = a0 * a0;
    F[56 + o] = 0.f;
    t = 0.f;
    for (int i = 1; i < 4; ++i) { float x = f2v[o * 13 + i]; t += x * x; }
    F[64 + o] = t * s1;
    F[72 + o] = 0.f;
    t = 0.f;
    for (int i = 4; i < 13; ++i) { float x = f2v[o * 13 + i]; t += x * x; }
    F[80 + o] = t * s2;
    F[88 + o] = 0.f;
  }
  float invacc[32];
  for (int o = 0; o < 32; ++o) invacc[o] = 0.f;
  const float scaleL[3] = {1.f, s1, s2};
  const int lo_[3] = {0, 1, 4}, hi_[3] = {1, 4, 13};
  const float* Winv[3] = {Winv0, Winv1, Winv2};
  for (int L = 0; L < 3; ++L) {
    for (int cc = 0; cc < 16; ++cc) {
      for (int dd = 0; dd < 16; ++dd) {
        float g = 0.f;
        for (int i = lo_[L]; i < hi_[L]; ++i) {
          float xc = (cc < 8) ? f1v[cc * 13 + i] : f2v[(cc - 8) * 13 + i];
          float xd = (dd < 8) ? f1v[dd * 13 + i] : f2v[(dd - 8) * 13 + i];
          g += xc * xd;
        }
        g *= scaleL[L];
        const float* W = Winv[L] + cc * 16 + dd;
#pragma unroll
        for (int o = 0; o < 32; ++o) invacc[o] += W[o * 256] * g;
      }
    }
  }
  for (int o = 0; o < 32; ++o) F[96 + o] = invacc[o];
}

// ---------------------------------------------------------------------------
// K6b: proj MLP (128->128 silu x3) via WMMA, final layer folded with ws_w
// into a 128-matvec, atomic per-graph accumulation.  One wave per 16 nodes.
// ---------------------------------------------------------------------------
__global__ __launch_bounds__(128) void k6b_proj(
    const float* __restrict__ feats, const float* __restrict__ p0w,
    const float* __restrict__ p0b, const float* __restrict__ p1w,
    const float* __restrict__ p1b, const float* __restrict__ p2w,
    const float* __restrict__ p2b, const float* __restrict__ p3w,
    const float* __restrict__ p3b, const float* __restrict__ wsw,
    const int* __restrict__ batchArr, float* __restrict__ outp, int N) {
  __shared__ __align__(16) float sm[4 * 2048];
  const int wid = threadIdx.x >> 5;
  const int lane = threadIdx.x & 31;
  const int tile = blockIdx.x * 4 + wid;
  const int n0 = tile * 16;
  if (n0 >= N) return;
  float* F = sm + wid * 2048;
  for (int idx = lane; idx < 2048; idx += 32) {
    int r = idx >> 7, k = idx & 127;
    F[idx] = (n0 + r < N) ? feats[(n0 + r) * 128 + k] : 0.f;
  }
  lds_wait();
  const int m = lane & 15;
  const int khi = (lane >= 16) ? 2 : 0;
  const int mhi = (lane >= 16) ? 8 : 0;
  const int ncol = lane & 15;
  const float* Ws[3] = {p0w, p1w, p2w};
  const float* Bs[3] = {p0b, p1b, p2b};
  for (int layer = 0; layer < 3; ++layer) {
    const float* W = Ws[layer];
    const float* Bb = Bs[layer];
    v8f acc[8];
    for (int nt = 0; nt < 8; ++nt) {
      v8f a8 = {};
      const int col = nt * 16 + ncol;
#pragma unroll
      for (int kk = 0; kk < 32; ++kk) {
        const int k0 = 4 * kk + khi;
        v2f a = *(const v2f*)(F + m * 128 + k0);
        v2f b;
        b.x = W[k0 * 128 + col];
        b.y = W[(k0 + 1) * 128 + col];
        a8 = wmma4(a, b, a8);
      }
      acc[nt] = a8;
    }
    lds_wait();
    for (int nt = 0; nt < 8; ++nt) {
      const int col = nt * 16 + ncol;
      float bb = Bb[col];
#pragma unroll
      for (int j = 0; j < 8; ++j)
        F[(j + mhi) * 128 + col] = silu_f(acc[nt][j] + bb);
    }
    lds_wait();
  }
  // node_e = bconst + sum_k F[r][k] * (sum_o p3w[k][o]*ws[o])
  if (lane < 16 && (n0 + lane) < N) {
    float acc = 0.f;
    for (int k = 0; k < 128; ++k) {
      float wc = 0.f;
      for (int o = 0; o < 17; ++o) wc += p3w[k * 17 + o] * wsw[o];
      acc += F[lane * 128 + k] * wc;
    }
    float bconst = 0.f;
    for (int o = 0; o < 17; ++o) bconst += p3b[o] * wsw[o];
    acc += bconst;
    atomicAdd(&outp[batchArr[n0 + lane]], acc);
  }
}

// ---------------------------------------------------------------------------
extern "C" void kernel_launch(void* const* d_in, const int* in_sizes, int n_in,
                              void* d_out, int out_size, void* d_ws,
                              size_t ws_size, hipStream_t stream) {
  // pytree-leaf order of setup_inputs()
  const float* pos = (const float*)d_in[0];
  const float* eshift = (const float*)d_in[1];
  const float* cell = (const float*)d_in[2];
  const float* emb_table = (const float*)d_in[3];
  const float* am0w = (const float*)d_in[4];
  const float* am0b = (const float*)d_in[5];
  const float* am1w = (const float*)d_in[6];
  const float* am1b = (const float*)d_in[7];
  const float* f1w0 = (const float*)d_in[8];
  const float* f1b0 = (const float*)d_in[9];
  const float* f1w1 = (const float*)d_in[10];
  const float* f1b1 = (const float*)d_in[11];
  const float* f1w2 = (const float*)d_in[12];
  const float* f1b2 = (const float*)d_in[13];
  const float* f2w0 = (const float*)d_in[14];
  const float* f2b0 = (const float*)d_in[15];
  const float* f2w1 = (const float*)d_in[16];
  const float* f2b1 = (const float*)d_in[17];
  const float* f2w2 = (const float*)d_in[18];
  const float* f2b2 = (const float*)d_in[19];
  const float* Winv0 = (const float*)d_in[20];
  const float* Winv1 = (const float*)d_in[21];
  const float* Winv2 = (const float*)d_in[22];
  const float* p0w = (const float*)d_in[23];
  const float* p0b = (const float*)d_in[24];
  const float* p1w = (const float*)d_in[25];
  const float* p1b = (const float*)d_in[26];
  const float* p2w = (const float*)d_in[27];
  const float* p2b = (const float*)d_in[28];
  const float* p3w = (const float*)d_in[29];
  const float* p3b = (const float*)d_in[30];
  const float* wsw = (const float*)d_in[31];
  const int* Aarr = (const int*)d_in[32];
  const int* batchArr = (const int*)d_in[33];
  const int* esrc = (const int*)d_in[34];
  const int* edst = (const int*)d_in[35];
  (void)n_in;

  const int N = in_sizes[32];
  const int E = in_sizes[34];

  float* ws = (float*)d_ws;
  size_t off = 0;
  float* ubuf = ws + off;   off += (size_t)E * 3;
  float* e2buf = ws + off;  off += (size_t)E * 9;
  float* embbuf = ws + off; off += (size_t)E * 16;
  float* Ai = ws + off;     off += (size_t)N * 8;
  float* deg = ws + off;    off += (size_t)N;       // deg | f1 | f2 contiguous
  float* f1 = ws + off;     off += (size_t)N * 104;
  float* f2 = ws + off;     off += (size_t)N * 104;
  float* feats = ws + off;  off += (size_t)N * 128;
  if (off * sizeof(float) > ws_size) return;  // insufficient scratch

  hipMemsetAsync(deg, 0, (size_t)(N + 208 * (size_t)N) * sizeof(float), stream);
  hipMemsetAsync(d_out, 0, (size_t)out_size * sizeof(float), stream);

  k1_geom<<<(E + 255) / 256, 256, 0, stream>>>(pos, eshift, cell, batchArr,
                                               esrc, edst, ubuf, e2buf, embbuf,
                                               deg, E);
  k2_atom<<<(N + 255) / 256, 256, 0, stream>>>(emb_table, Aarr, am0w, am0b,
                                               am1w, am1b, Ai, N);
  const int tilesE = (E + 15) / 16;
  k3_fc1<<<(tilesE + 3) / 4, 128, 0, stream>>>(embbuf, Ai, esrc, edst, ubuf,
                                               e2buf, f1w0, f1b0, f1w1, f1b1,
                                               f1w2, f1b2, f1, E);
  k4_mean<<<((N * 104) + 255) / 256, 256, 0, stream>>>(f1, deg, N * 104);
  k5_fc2<<<(tilesE + 1) / 2, 64, 0, stream>>>(embbuf, f1, esrc, edst, ubuf,
                                              e2buf, f2w0, f2b0, f2w1, f2b1,
                                              f2w2, f2b2, f2, E);
  k6a_feats<<<(N + 255) / 256, 256, 0, stream>>>(f1, f2, deg, Winv0, Winv1,
                                                 Winv2, feats, N);
  const int tilesN = (N + 15) / 16;
  k6b_proj<<<(tilesN + 3) / 4, 128, 0, stream>>>(feats, p0w, p0b, p1w, p1b,
                                                 p2w, p2b, p3w, p3b, wsw,
                                                 batchArr, (float*)d_out, N);
}